// GNN_42571715838145
// MI455X (gfx1250) — compile-verified
//
#include <hip/hip_runtime.h>

#define HID 128
#define INDIM 64

typedef __attribute__((ext_vector_type(2))) float v2f;
typedef __attribute__((ext_vector_type(8))) float v8f;

// ---------------------------------------------------------------------------
// WMMA f32 GEMM: H[nrows x 128] = X[nrows x K] @ W[K x 128]
// block = 256 threads = 8 waves; each wave -> one 16x16 tile; block -> 16 rows
// W staged in dynamic LDS (K*128*4 bytes).
// ---------------------------------------------------------------------------
template <int K>
__global__ void gcn_gemm_wmma(const float* __restrict__ X,
                              const float* __restrict__ W,
                              float* __restrict__ H, int nrows) {
  extern __shared__ float sW[];
  const int tid = threadIdx.x;
  for (int i = tid; i < K * HID; i += 256) sW[i] = W[i];
  __syncthreads();

  const int m0 = blockIdx.x * 16;
  if (m0 >= nrows) return;                 // uniform per block: EXEC stays full
  const int lane = tid & 31;
  const int wave = tid >> 5;               // 0..7 -> column tile
  const int n0   = wave * 16;
  const int half = lane >> 4;              // 0: lanes 0-15, 1: lanes 16-31
  const int l15  = lane & 15;
  const int kp   = half * 2;               // K sub-offset held by this half-wave

  int arow = m0 + l15;
  if (arow >= nrows) arow = nrows - 1;     // clamp loads (stores guarded below)
  const float* xrow = X + (size_t)arow * K;

  v8f acc = {};
#pragma unroll 4
  for (int k0 = 0; k0 < K; k0 += 4) {
    v2f a, b;
    a[0] = xrow[k0 + kp];
    a[1] = xrow[k0 + kp + 1];
    b[0] = sW[(k0 + kp) * HID + n0 + l15];
    b[1] = sW[(k0 + kp + 1) * HID + n0 + l15];
    acc = __builtin_amdgcn_wmma_f32_16x16x4_f32(
        /*neg_a=*/false, a, /*neg_b=*/false, b,
        /*c_mod=*/(short)0, acc, /*reuse_a=*/false, /*reuse_b=*/false);
  }

  const int mbase = m0 + half * 8;
  const int ncol  = n0 + l15;
  if (m0 + 16 <= nrows) {
    // fast path: full tile, unguarded coalesced stores (no exec toggling)
#pragma unroll
    for (int r = 0; r < 8; ++r)
      H[(size_t)(mbase + r) * HID + ncol] = acc[r];
  } else {
#pragma unroll
    for (int r = 0; r < 8; ++r) {
      const int mr = mbase + r;
      if (mr < nrows) H[(size_t)mr * HID + ncol] = acc[r];
    }
  }
}

// ---------------------------------------------------------------------------
// Degree / normalization
// ---------------------------------------------------------------------------
__global__ void deg_init(float* __restrict__ deg, int n) {
  int i = blockIdx.x * 256 + threadIdx.x;
  if (i < n) deg[i] = 1.0f;                // self loop
}
__global__ void deg_count(const int* __restrict__ dst, float* __restrict__ deg,
                          int ne) {
  int e = blockIdx.x * 256 + threadIdx.x;
  if (e < ne) atomicAdd(&deg[dst[e]], 1.0f);
}
__global__ void deg_rsqrt(float* __restrict__ dinv, int n) {
  int i = blockIdx.x * 256 + threadIdx.x;
  if (i < n) dinv[i] = rsqrtf(dinv[i]);
}

// ---------------------------------------------------------------------------
// AGG init: agg[i][c] = h[i][c] * dinv[i]^2 + b[c]   (self-loop + bias fused)
// ---------------------------------------------------------------------------
__global__ void agg_init(const float* __restrict__ H,
                         const float* __restrict__ dinv,
                         const float* __restrict__ bias,
                         float* __restrict__ AGG, int nelem) {
  int idx = blockIdx.x * 256 + threadIdx.x;
  if (idx >= nelem) return;
  int node = idx >> 7;         // /128
  int c    = idx & (HID - 1);
  float di = dinv[node];
  AGG[idx] = H[idx] * di * di + bias[c];
}

// ---------------------------------------------------------------------------
// Edge scatter: one wave per edge; lane l handles cols [4l, 4l+4)
// agg[dst] += h[src] * dinv[src]*dinv[dst]
// Edge index is wave-uniform: force it scalar so src/dst/dinv go through
// SMEM/SGPRs instead of 32 redundant per-lane vector loads.
// ---------------------------------------------------------------------------
__global__ void edge_scatter(const float* __restrict__ H,
                             const int* __restrict__ src,
                             const int* __restrict__ dst,
                             const float* __restrict__ dinv,
                             float* __restrict__ AGG, int ne) {
  int e = __builtin_amdgcn_readfirstlane(blockIdx.x * 8 + (threadIdx.x >> 5));
  if (e >= ne) return;
  int lane = threadIdx.x & 31;
  int s = __builtin_amdgcn_readfirstlane(src[e]);
  int d = __builtin_amdgcn_readfirstlane(dst[e]);
  float nrm = dinv[s] * dinv[d];
  float4 v = ((const float4*)(H + (size_t)s * HID))[lane];
  float* a = AGG + (size_t)d * HID + lane * 4;
  atomicAdd(a + 0, v.x * nrm);
  atomicAdd(a + 1, v.y * nrm);
  atomicAdd(a + 2, v.z * nrm);
  atomicAdd(a + 3, v.w * nrm);
}

__global__ void relu_inplace(float* __restrict__ A, int nelem) {
  int idx = blockIdx.x * 256 + threadIdx.x;
  if (idx < nelem) A[idx] = fmaxf(A[idx], 0.0f);
}

// ---------------------------------------------------------------------------
// Mean pool + final linear
// ---------------------------------------------------------------------------
__global__ void pool_zero(float* __restrict__ pool, float* __restrict__ cnt,
                          int g) {
  int idx = blockIdx.x * 256 + threadIdx.x;
  if (idx < g * HID) pool[idx] = 0.0f;
  if (idx < g) cnt[idx] = 0.0f;
}
__global__ void pool_scatter(const float* __restrict__ H,
                             const int* __restrict__ batch,
                             float* __restrict__ pool,
                             float* __restrict__ cnt, int n) {
  int node = __builtin_amdgcn_readfirstlane(blockIdx.x * 8 + (threadIdx.x >> 5));
  if (node >= n) return;
  int lane = threadIdx.x & 31;
  int g = __builtin_amdgcn_readfirstlane(batch[node]);
  float4 v = ((const float4*)(H + (size_t)node * HID))[lane];
  float* p = pool + (size_t)g * HID + lane * 4;
  atomicAdd(p + 0, v.x);
  atomicAdd(p + 1, v.y);
  atomicAdd(p + 2, v.z);
  atomicAdd(p + 3, v.w);
  if (lane == 0) atomicAdd(&cnt[g], 1.0f);
}
__global__ void final_linear(const float* __restrict__ pool,
                             const float* __restrict__ cnt,
                             const float* __restrict__ Wl,
                             const float* __restrict__ bl,
                             float* __restrict__ out, int ngraphs) {
  int g = __builtin_amdgcn_readfirstlane(blockIdx.x * 8 + (threadIdx.x >> 5));
  if (g >= ngraphs) return;
  int lane = threadIdx.x & 31;
  float s = 0.0f;
#pragma unroll
  for (int c = lane; c < HID; c += 32) s += pool[(size_t)g * HID + c] * Wl[c];
#pragma unroll
  for (int off = 16; off > 0; off >>= 1) s += __shfl_down(s, off, 32);
  if (lane == 0) out[g] = s / fmaxf(cnt[g], 1.0f) + bl[0];
}

// ---------------------------------------------------------------------------
extern "C" void kernel_launch(void* const* d_in, const int* in_sizes, int n_in,
                              void* d_out, int out_size, void* d_ws,
                              size_t ws_size, hipStream_t stream) {
  const float* x     = (const float*)d_in[0];
  const int*   src   = (const int*)d_in[1];
  const int*   dst   = (const int*)d_in[2];
  const int*   batch = (const int*)d_in[3];
  const float* W1 = (const float*)d_in[4];
  const float* b1 = (const float*)d_in[5];
  const float* W2 = (const float*)d_in[6];
  const float* b2 = (const float*)d_in[7];
  const float* W3 = (const float*)d_in[8];
  const float* b3 = (const float*)d_in[9];
  const float* Wl = (const float*)d_in[10];
  const float* bl = (const float*)d_in[11];
  float* out = (float*)d_out;

  const int N  = in_sizes[0] / INDIM;   // 100000
  const int NE = in_sizes[1];           // 1600000
  const int G  = out_size;              // 1000 (OUT==1)
  const int nelem = N * HID;

  float* bufA = (float*)d_ws;                       // [N,128]
  float* bufB = bufA + (size_t)N * HID;             // [N,128]
  float* dinv = bufB + (size_t)N * HID;             // [N] (deg then rsqrt)
  float* pool = dinv + N;                           // [G,128]
  float* cnt  = pool + (size_t)G * HID;             // [G]

  dim3 blk(256);
  const int gElem = (nelem + 255) / 256;
  const int gNode = (N + 255) / 256;
  const int gEdge = (NE + 255) / 256;
  const int gEdgeW = (NE + 7) / 8;   // wave per edge
  const int gRows = (N + 15) / 16;   // 16 rows per block
  const size_t lds64  = (size_t)INDIM * HID * sizeof(float);  // 32 KB
  const size_t lds128 = (size_t)HID * HID * sizeof(float);    // 64 KB

  // degrees -> dinv
  deg_init<<<gNode, blk, 0, stream>>>(dinv, N);
  deg_count<<<gEdge, blk, 0, stream>>>(dst, dinv, NE);
  deg_rsqrt<<<gNode, blk, 0, stream>>>(dinv, N);

  // ---- layer 1: h = relu(gcn(x, W1, b1)) ----
  gcn_gemm_wmma<INDIM><<<gRows, blk, lds64, stream>>>(x, W1, bufA, N);
  agg_init<<<gElem, blk, 0, stream>>>(bufA, dinv, b1, bufB, nelem);
  edge_scatter<<<gEdgeW, blk, 0, stream>>>(bufA, src, dst, dinv, bufB, NE);
  relu_inplace<<<gElem, blk, 0, stream>>>(bufB, nelem);

  // ---- layer 2 ----
  gcn_gemm_wmma<HID><<<gRows, blk, lds128, stream>>>(bufB, W2, bufA, N);
  agg_init<<<gElem, blk, 0, stream>>>(bufA, dinv, b2, bufB, nelem);
  edge_scatter<<<gEdgeW, blk, 0, stream>>>(bufA, src, dst, dinv, bufB, NE);
  relu_inplace<<<gElem, blk, 0, stream>>>(bufB, nelem);

  // ---- layer 3 (no relu) ----
  gcn_gemm_wmma<HID><<<gRows, blk, lds128, stream>>>(bufB, W3, bufA, N);
  agg_init<<<gElem, blk, 0, stream>>>(bufA, dinv, b3, bufB, nelem);
  edge_scatter<<<gEdgeW, blk, 0, stream>>>(bufA, src, dst, dinv, bufB, NE);

  // ---- mean pool + linear head ----
  pool_zero<<<(G * HID + 255) / 256, blk, 0, stream>>>(pool, cnt, G);
  pool_scatter<<<(N + 7) / 8, blk, 0, stream>>>(bufB, batch, pool, cnt, N);
  final_linear<<<(G + 7) / 8, blk, 0, stream>>>(pool, cnt, Wl, bl, out, G);
}